// SetOfOrthonormalTransforms_72490458022405
// MI455X (gfx1250) — compile-verified
//
#include <hip/hip_runtime.h>

typedef __attribute__((ext_vector_type(2))) float v2f;
typedef __attribute__((ext_vector_type(4))) float v4f;
typedef __attribute__((ext_vector_type(8))) float v8f;

#define NDIM 16
#define NANG 120   // 16*15/2

// One workgroup per block (nblks total). 256 threads = 8 wave32.
// Phase 1: lanes 0..15 build R (Givens cascade) -> LDS (column-major).
// Phase 2: all 8 waves compute R @ X over 16-column tiles with
//          V_WMMA_F32_16X16X4_F32 (4 chunks of K=4).
// All in-block offsets are unsigned 32-bit: loads/stores use SADDR+voffset.
__global__ __launch_bounds__(256) void orth_wmma_kernel(
    const float* __restrict__ X,
    const float* __restrict__ angles,
    const float* __restrict__ mus,
    float* __restrict__ out,
    int m)
{
    __shared__ float Rcm[NDIM * NDIM];   // Rcm[j*16 + i] = R[i][j] (column-major)

    const int blk = blockIdx.x;
    const int tid = threadIdx.x;

    // ---------------- Phase 1: build R (16 lanes, one column each) -------------
    if (tid < NDIM) {
        const int j = tid;
        float mreg[NDIM];
#pragma unroll
        for (int i = 0; i < NDIM; ++i) mreg[i] = (i == j) ? 1.0f : 0.0f;

        const float* ang = angles + (size_t)blk * NANG;
        int k = 0;
#pragma unroll
        for (int iTop = 0; iTop < NDIM - 1; ++iTop) {
#pragma unroll
            for (int iBtm = iTop + 1; iBtm < NDIM; ++iBtm) {
                const float a = ang[k++];          // wave-uniform -> s_load
                const float c = __cosf(a);
                const float s = __sinf(a);
                const float t = c * mreg[iTop] - s * mreg[iBtm];
                mreg[iBtm]    = s * mreg[iTop] + c * mreg[iBtm];
                mreg[iTop]    = t;
            }
        }
        const float* mu = mus + (size_t)blk * NDIM;
#pragma unroll
        for (int i = 0; i < NDIM; ++i) mreg[i] *= mu[i];

        // column j is contiguous in column-major layout -> 4x ds_store_b128
#pragma unroll
        for (int i = 0; i < NDIM; i += 4) {
            v4f v = { mreg[i], mreg[i + 1], mreg[i + 2], mreg[i + 3] };
            *(v4f*)&Rcm[j * NDIM + i] = v;
        }
    }
    __syncthreads();

    // ---------------- Phase 2: WMMA R @ X over column tiles --------------------
    const int lane = tid & 31;
    const int wave = tid >> 5;
    const int half = lane >> 4;    // which 16-lane half of the wave
    const int mrow = lane & 15;    // M index (A/D) == N index (B/D)

    // A operand: 16x4 f32 chunk of R. Lane L, VGPR v holds
    //   A[M = L%16][K = 4*kc + 2*(L/16) + v]
    v2f a[4];
#pragma unroll
    for (int kc = 0; kc < 4; ++kc) {
        const int k0 = 4 * kc + 2 * half;
        a[kc].x = Rcm[(k0    ) * NDIM + mrow];
        a[kc].y = Rcm[(k0 + 1) * NDIM + mrow];
    }

    // Uniform 64-bit bases (SGPRs); per-lane offsets stay unsigned 32-bit.
    const float* __restrict__ Xb = X   + (size_t)blk * NDIM * (size_t)m;
    float* __restrict__       Ob = out + (size_t)blk * NDIM * (size_t)m;

    const unsigned um     = (unsigned)m;
    const unsigned ntiles = um >> 4;           // 16 columns per tile

    // Loop-invariant row bases (element offsets), 8 for loads, 8 for stores.
    unsigned lrow[8];
#pragma unroll
    for (int kc = 0; kc < 4; ++kc) {
#pragma unroll
        for (int v = 0; v < 2; ++v) {
            lrow[2 * kc + v] = (unsigned)(4 * kc + 2 * half + v) * um;
        }
    }
    unsigned orow[8];
#pragma unroll
    for (int p = 0; p < 8; ++p) {
        orow[p] = (unsigned)(p + 8 * half) * um;
    }

    // per-lane running column offset for this wave
    unsigned cb = (unsigned)(wave * 16 + mrow);
    const unsigned cstep = 8 * 16;             // 8 waves x 16 columns

    for (unsigned t = (unsigned)wave; t < ntiles; t += 8, cb += cstep) {
        // prefetch the tile this wave touches next iteration (global_prefetch_b8)
        if (t + 8 < ntiles) {
            __builtin_prefetch(&Xb[lrow[0] + cb + cstep], 0, 1);
        }

        v8f acc = {};
#pragma unroll
        for (int kc = 0; kc < 4; ++kc) {
            // B operand: 4x16 f32 chunk of X. Lane L, VGPR v holds
            //   B[K = 4*kc + 2*(L/16) + v][N = L%16]
            v2f b;
            b.x = Xb[lrow[2 * kc    ] + cb];
            b.y = Xb[lrow[2 * kc + 1] + cb];
            // 8 args: (neg_a, A, neg_b, B, c_mod, C, reuse_a, reuse_b)
            acc = __builtin_amdgcn_wmma_f32_16x16x4_f32(
                false, a[kc], false, b, (short)0, acc, false, false);
        }

        // D layout: VGPR p holds D[M = p + 8*(L/16)][N = L%16]
#pragma unroll
        for (int p = 0; p < 8; ++p) {
            Ob[orow[p] + cb] = acc[p];
        }
    }
}

extern "C" void kernel_launch(void* const* d_in, const int* in_sizes, int n_in,
                              void* d_out, int out_size, void* d_ws, size_t ws_size,
                              hipStream_t stream) {
    const float* X      = (const float*)d_in[0];  // (nblks, 16, m) f32
    const float* angles = (const float*)d_in[1];  // (nblks, 120) f32
    const float* mus    = (const float*)d_in[2];  // (nblks, 16)  f32
    float* out          = (float*)d_out;          // (nblks, 16, m) f32

    const int nblks = in_sizes[2] / NDIM;
    const int m     = (int)((size_t)in_sizes[0] / ((size_t)nblks * NDIM));

    orth_wmma_kernel<<<nblks, 256, 0, stream>>>(X, angles, mus, out, m);
}